// Net_18399639896776
// MI455X (gfx1250) — compile-verified
//
#include <hip/hip_runtime.h>
#include <hip/hip_bf16.h>
#include <math.h>

typedef float v2f __attribute__((ext_vector_type(2)));
typedef float v8f __attribute__((ext_vector_type(8)));

#define F_IN 512
#define HDIM 16
#define CDIM 40

// ---------------- degree + rsqrt ----------------
__global__ void deg_kernel(const int* __restrict__ cols, const float* __restrict__ w,
                           float* __restrict__ deg, int ne) {
    int e = blockIdx.x * blockDim.x + threadIdx.x;
    if (e >= ne) return;
    unsafeAtomicAdd(&deg[cols[e]], w[e]);
}

__global__ void dinv_kernel(float* __restrict__ deg, int n) {
    int i = blockIdx.x * blockDim.x + threadIdx.x;
    if (i >= n) return;
    float d = deg[i];
    deg[i] = (d > 0.0f) ? rsqrtf(d) : 0.0f;
}

// ---------------- per-edge norm = dinv[row] * w * dinv[col] ----------------
__global__ void norm_kernel(const int* __restrict__ rows, const int* __restrict__ cols,
                            const float* __restrict__ w, const float* __restrict__ dinv,
                            float* __restrict__ norm, int ne) {
    int e = blockIdx.x * blockDim.x + threadIdx.x;
    if (e >= ne) return;
    norm[e] = dinv[rows[e]] * w[e] * dinv[cols[e]];
}

// ---------------- xw = x @ W1 via V_WMMA_F32_16X16X4_F32 ----------------
// One wave per 16-row tile of x. A: 16x4 f32 (lanes 0-15: K=0,1 ; lanes 16-31: K=2,3).
// W1 staged in LDS pair-interleaved: pair p=k/2 holds {W1[2p][n], W1[2p+1][n]} at
// float2 index p*16+n  ->  one conflict-free ds_load_b64 per lane per K-step.
__global__ void gcn_xw_wmma(const float* __restrict__ x, const float* __restrict__ W1,
                            float* __restrict__ xw, int nrows) {
    __shared__ float sW[F_IN * HDIM];   // 32 KB
    for (int idx = threadIdx.x; idx < F_IN * HDIM; idx += blockDim.x) {
        int k = idx >> 4, n = idx & 15;
        sW[(k >> 1) * 32 + n * 2 + (k & 1)] = W1[idx];
    }
    __syncthreads();

    const int lane = threadIdx.x & 31;
    const int wib  = threadIdx.x >> 5;
    const int wpb  = blockDim.x >> 5;
    const int tile = blockIdx.x * wpb + wib;
    const int row0 = tile * 16;
    if (row0 >= nrows) return;              // wave-uniform: EXEC stays all-ones for WMMA

    const int half = lane >> 4;             // 0: K+0/1 ; 1: K+2/3
    const int l16  = lane & 15;
    int rowA = row0 + l16;
    if (rowA >= nrows) rowA = nrows - 1;    // clamp loads; stores guarded below
    const float* __restrict__ xrow = x + (size_t)rowA * F_IN + half * 2;
    const v2f* __restrict__ wp = (const v2f*)sW;

    v8f acc = {};
    for (int k = 0; k < F_IN; k += 4) {
        v2f a = *(const v2f*)(xrow + k);                 // global_load_b64
        v2f b = wp[((k >> 1) + half) * 16 + l16];        // ds_load_b64
        acc = __builtin_amdgcn_wmma_f32_16x16x4_f32(
            false, a, false, b, (short)0, acc, false, false);
    }

    // D layout: VGPR v, lane l16 = column n; row m = v + half*8
    float* __restrict__ o = xw + ((size_t)row0 + half * 8) * HDIM + l16;
    if (row0 + 16 <= nrows) {               // full tile fast path (always, N % 16 == 0)
#pragma unroll
        for (int v = 0; v < 8; ++v) o[(size_t)v * HDIM] = acc[v];
    } else {
#pragma unroll
        for (int v = 0; v < 8; ++v) {
            int r = row0 + v + half * 8;
            if (r < nrows) xw[(size_t)r * HDIM + l16] = acc[v];
        }
    }
}

// ---------------- edge aggregation: dst[col] += src[row] * norm[e] ----------------
__global__ void gcn_aggregate16(const float* __restrict__ src,
                                const int* __restrict__ rows, const int* __restrict__ cols,
                                const float* __restrict__ norm,
                                float* __restrict__ dst, int ne) {
    int e = blockIdx.x * blockDim.x + threadIdx.x;
    if (e >= ne) return;
    int r = rows[e], c = cols[e];
    float nm = norm[e];
    const float4* __restrict__ s = (const float4*)(src + (size_t)r * HDIM);
    float* __restrict__ d = dst + (size_t)c * HDIM;
#pragma unroll
    for (int q = 0; q < 4; ++q) {
        float4 v = s[q];
        unsafeAtomicAdd(d + q * 4 + 0, v.x * nm);
        unsafeAtomicAdd(d + q * 4 + 1, v.y * nm);
        unsafeAtomicAdd(d + q * 4 + 2, v.z * nm);
        unsafeAtomicAdd(d + q * 4 + 3, v.w * nm);
    }
}

// ---------------- h = relu(agg + b1), in place ----------------
__global__ void relu_bias_kernel(float* __restrict__ h, const float* __restrict__ b1, int total) {
    int i = blockIdx.x * blockDim.x + threadIdx.x;
    if (i >= total) return;
    float v = h[i] + b1[i & (HDIM - 1)];
    h[i] = v > 0.0f ? v : 0.0f;
}

// ---------------- out = log_softmax(agg2 @ W2 + b2) ----------------
__global__ void gcn_head(const float* __restrict__ agg, const float* __restrict__ W2,
                         const float* __restrict__ b2, float* __restrict__ out, int n) {
    __shared__ float sW[HDIM * CDIM];
    __shared__ float sb[CDIM];
    for (int i = threadIdx.x; i < HDIM * CDIM; i += blockDim.x) sW[i] = W2[i];
    if (threadIdx.x < CDIM) sb[threadIdx.x] = b2[threadIdx.x];
    __syncthreads();

    int i = blockIdx.x * blockDim.x + threadIdx.x;
    if (i >= n) return;

    float a[HDIM];
    const float4* __restrict__ p = (const float4*)(agg + (size_t)i * HDIM);
#pragma unroll
    for (int q = 0; q < 4; ++q) {
        float4 v = p[q];
        a[q * 4 + 0] = v.x; a[q * 4 + 1] = v.y; a[q * 4 + 2] = v.z; a[q * 4 + 3] = v.w;
    }

    float v[CDIM];
    float m = -INFINITY;
#pragma unroll
    for (int c = 0; c < CDIM; ++c) {
        float acc = sb[c];
#pragma unroll
        for (int f = 0; f < HDIM; ++f) acc += a[f] * sW[f * CDIM + c];
        v[c] = acc;
        m = fmaxf(m, acc);
    }
    float se = 0.0f;
#pragma unroll
    for (int c = 0; c < CDIM; ++c) se += __expf(v[c] - m);
    float lse = m + __logf(se);
    float* __restrict__ o = out + (size_t)i * CDIM;
#pragma unroll
    for (int c = 0; c < CDIM; ++c) o[c] = v[c] - lse;
}

extern "C" void kernel_launch(void* const* d_in, const int* in_sizes, int n_in,
                              void* d_out, int out_size, void* d_ws, size_t ws_size,
                              hipStream_t stream) {
    const float* x  = (const float*)d_in[0];
    const int*   ei = (const int*)d_in[1];
    const float* ew = (const float*)d_in[2];
    const float* W1 = (const float*)d_in[3];
    const float* b1 = (const float*)d_in[4];
    const float* W2 = (const float*)d_in[5];
    const float* b2 = (const float*)d_in[6];
    float* out = (float*)d_out;

    const int N  = in_sizes[0] / F_IN;   // 100000
    const int NE = in_sizes[2];          // E + N
    const int* rows = ei;                // edge_index[0]
    const int* cols = ei + NE;           // edge_index[1]

    float* deg  = (float*)d_ws;                    // N   (becomes dinv in place)
    float* nrm  = deg + N;                         // NE
    float* buf1 = nrm + NE;                        // 16N (xw, later agg2)
    float* buf2 = buf1 + (size_t)N * HDIM;         // 16N (agg1 -> h)

    hipMemsetAsync(deg,  0, (size_t)N * sizeof(float), stream);
    hipMemsetAsync(buf2, 0, (size_t)N * HDIM * sizeof(float), stream);

    const int TB = 256;
    deg_kernel<<<(NE + TB - 1) / TB, TB, 0, stream>>>(cols, ew, deg, NE);
    dinv_kernel<<<(N + TB - 1) / TB, TB, 0, stream>>>(deg, N);
    norm_kernel<<<(NE + TB - 1) / TB, TB, 0, stream>>>(rows, cols, ew, deg, nrm, NE);

    const int tiles = (N + 15) / 16;
    gcn_xw_wmma<<<(tiles + 7) / 8, 256, 0, stream>>>(x, W1, buf1, N);

    gcn_aggregate16<<<(NE + TB - 1) / TB, TB, 0, stream>>>(buf1, rows, cols, nrm, buf2, NE);
    relu_bias_kernel<<<((N * HDIM) + TB - 1) / TB, TB, 0, stream>>>(buf2, b1, N * HDIM);

    hipMemsetAsync(buf1, 0, (size_t)N * HDIM * sizeof(float), stream);
    gcn_aggregate16<<<(NE + TB - 1) / TB, TB, 0, stream>>>(buf2, rows, cols, nrm, buf1, NE);

    gcn_head<<<(N + TB - 1) / TB, TB, 0, stream>>>(buf1, W2, b2, out, N);
}